// QEffDeepseekV3Attention_86509231276098
// MI455X (gfx1250) — compile-verified
//
#include <hip/hip_runtime.h>
#include <hip/hip_bf16.h>
#include <math.h>

// ---------------------------------------------------------------------------
// DeepSeek-V3 MLA attention (weight-absorbed), fp32, on gfx1250 via
// V_WMMA_F32_16X16X4_F32 (full fp32 precision on the matrix pipe).
// ---------------------------------------------------------------------------

typedef float v2f __attribute__((ext_vector_type(2)));
typedef float v8f __attribute__((ext_vector_type(8)));

#define S_LEN   2048
#define HID     7168
#define NH      32
#define Q_LORA  1536
#define KV_LORA 512
#define ROPE_D  64
#define CDIM    576          // KV_LORA + ROPE_D
#define V_D     128
#define OUTHW   4096         // NH * V_D
#define SM_SCALE 0.07216878364870323f   // 1/sqrt(192)
#define MIN_MASKED -10000.0f

static __device__ __forceinline__ v8f wmma_f32(v2f a, v2f b, v8f c) {
    return __builtin_amdgcn_wmma_f32_16x16x4_f32(false, a, false, b,
                                                 (short)0, c, false, false);
}

static __device__ __forceinline__ v8f v8f_zero() {
    v8f z = {0.f,0.f,0.f,0.f,0.f,0.f,0.f,0.f};
    return z;
}

// 8-byte vector load (global or LDS pointer; all call sites are 8B-aligned)
static __device__ __forceinline__ v2f ld2(const float* p) {
    return *(const v2f*)p;
}

// ---------------------------------------------------------------------------
// Generic fp32 WMMA GEMM: C[M,N] = A[M,K] @ B[K,N].
// Block = 256 threads = 8 waves arranged 4(M) x 2(N); each wave owns a
// 16x64 strip (4 accumulators, A-fragment reused 4x per K-step).
// Branch-free K-loop: out-of-range column tiles are clamped to N-16 (loads
// stay in-bounds, redundant results discarded at store time).
// Requires M % 64 == 0, N % 16 == 0, N >= 16, K % 4 == 0.
// ---------------------------------------------------------------------------
__global__ __launch_bounds__(256) void gemm_wmma_f32(
    const float* __restrict__ A, const float* __restrict__ B,
    float* __restrict__ C, int M, int N, int K, int lda, int ldb, int ldc)
{
    const int tid   = threadIdx.x;
    const int w     = tid >> 5;
    const int lane  = tid & 31;
    const int half  = lane >> 4;
    const int khalf = 2 * half;
    const int l16   = lane & 15;
    const int wm    = w >> 1;
    const int wn    = w & 1;

    const int row0 = blockIdx.y * 64 + wm * 16;
    const int col0 = blockIdx.x * 128 + wn * 64;
    if (row0 >= M) return;

    int  cb[4];
    bool cv[4];
    for (int j = 0; j < 4; ++j) {
        const int c = col0 + j * 16;
        cv[j] = (c < N);
        cb[j] = cv[j] ? c : (N - 16);   // clamp: compute garbage, never store
    }

    v8f acc0 = v8f_zero(), acc1 = v8f_zero(), acc2 = v8f_zero(), acc3 = v8f_zero();

    const float* __restrict__ arow = A + (size_t)(row0 + l16) * lda + khalf;
    const float* __restrict__ b0p  = B + cb[0] + l16;
    const float* __restrict__ b1p  = B + cb[1] + l16;
    const float* __restrict__ b2p  = B + cb[2] + l16;
    const float* __restrict__ b3p  = B + cb[3] + l16;
    const size_t ldbs = (size_t)ldb;

    for (int k = 0; k < K; k += 4) {
        const v2f a = ld2(arow + k);
        const size_t ro = (size_t)(k + khalf) * ldbs;
        v2f b;
        b.x = b0p[ro]; b.y = b0p[ro + ldbs]; acc0 = wmma_f32(a, b, acc0);
        b.x = b1p[ro]; b.y = b1p[ro + ldbs]; acc1 = wmma_f32(a, b, acc1);
        b.x = b2p[ro]; b.y = b2p[ro + ldbs]; acc2 = wmma_f32(a, b, acc2);
        b.x = b3p[ro]; b.y = b3p[ro + ldbs]; acc3 = wmma_f32(a, b, acc3);
    }

    v8f accs[4] = {acc0, acc1, acc2, acc3};
    for (int j = 0; j < 4; ++j) {
        if (!cv[j]) continue;
        for (int r = 0; r < 8; ++r) {
            C[(size_t)(row0 + r + 8 * half) * ldc + cb[j] + l16] = accs[j][r];
        }
    }
}

// ---------------------------------------------------------------------------
// In-place RMSNorm over rows: x[row] = x[row] * rsqrt(mean(x^2)+eps) * w
// ---------------------------------------------------------------------------
__global__ __launch_bounds__(256) void rmsnorm_rows(
    float* __restrict__ x, const float* __restrict__ w, int d)
{
    const int s = blockIdx.x;
    float* row = x + (size_t)s * d;
    __shared__ float red[256];
    __shared__ float rinv;

    float ss = 0.f;
    for (int i = threadIdx.x; i < d; i += 256) { float v = row[i]; ss += v * v; }
    red[threadIdx.x] = ss;
    __syncthreads();
    for (int off = 128; off > 0; off >>= 1) {
        if (threadIdx.x < off) red[threadIdx.x] += red[threadIdx.x + off];
        __syncthreads();
    }
    if (threadIdx.x == 0) rinv = rsqrtf(red[0] / (float)d + 1e-6f);
    __syncthreads();
    for (int i = threadIdx.x; i < d; i += 256) row[i] = row[i] * rinv * w[i];
}

// ---------------------------------------------------------------------------
// Build k_full[s, 0:512] = rmsnorm(compressed[s, 0:512]) * kv_a_ln_w
//       k_full[s, 512:576] = RoPE(compressed[s, 512:576]) (deinterleave form)
// ---------------------------------------------------------------------------
__global__ __launch_bounds__(256) void build_kfull(
    const float* __restrict__ comp, const float* __restrict__ kvw,
    const int* __restrict__ pos_ids, float* __restrict__ kf)
{
    const int s = blockIdx.x;
    const float* c = comp + (size_t)s * CDIM;
    float* o = kf + (size_t)s * CDIM;
    __shared__ float red[256];
    __shared__ float rinv;

    float ss = 0.f;
    for (int i = threadIdx.x; i < KV_LORA; i += 256) { float v = c[i]; ss += v * v; }
    red[threadIdx.x] = ss;
    __syncthreads();
    for (int off = 128; off > 0; off >>= 1) {
        if (threadIdx.x < off) red[threadIdx.x] += red[threadIdx.x + off];
        __syncthreads();
    }
    if (threadIdx.x == 0) rinv = rsqrtf(red[0] / (float)KV_LORA + 1e-6f);
    __syncthreads();
    for (int i = threadIdx.x; i < KV_LORA; i += 256) o[i] = c[i] * rinv * kvw[i];

    if (threadIdx.x < 32) {
        const int i = threadIdx.x;
        const float p = (float)pos_ids[s];
        const float invf = powf(10000.0f, -(float)i / 32.0f);
        const float f = p * invf;
        const float cs = cosf(f), sn = sinf(f);
        const float x0 = c[KV_LORA + 2 * i];
        const float x1 = c[KV_LORA + 2 * i + 1];
        o[KV_LORA + i]      = x0 * cs - x1 * sn;   // even half
        o[KV_LORA + 32 + i] = x1 * cs + x0 * sn;   // odd half
    }
}

// ---------------------------------------------------------------------------
// In-place interleaved RoPE on q_pe[s, h*64 + d] for all 32 heads of row s.
// ---------------------------------------------------------------------------
__global__ __launch_bounds__(256) void rope_qpe(
    float* __restrict__ q, const int* __restrict__ pos_ids)
{
    const int s = blockIdx.x;
    const float p = (float)pos_ids[s];
    float x0v[4], x1v[4];
    for (int it = 0; it < 4; ++it) {
        const int idx = threadIdx.x + it * 256;   // 0..1023: h in [0,32), i in [0,32)
        const int hh = idx >> 5, i = idx & 31;
        const size_t base = (size_t)s * (NH * ROPE_D) + hh * ROPE_D;
        x0v[it] = q[base + 2 * i];
        x1v[it] = q[base + 2 * i + 1];
    }
    __syncthreads();
    for (int it = 0; it < 4; ++it) {
        const int idx = threadIdx.x + it * 256;
        const int hh = idx >> 5, i = idx & 31;
        const size_t base = (size_t)s * (NH * ROPE_D) + hh * ROPE_D;
        const float invf = powf(10000.0f, -(float)i / 32.0f);
        const float f = p * invf;
        const float cs = cosf(f), sn = sinf(f);
        q[base + i]      = x0v[it] * cs - x1v[it] * sn;
        q[base + 32 + i] = x1v[it] * cs + x0v[it] * sn;
    }
}

// ---------------------------------------------------------------------------
// Fused MLA flash attention per (head h, 16-row query tile).
//   q_full tile  = [ q_a@fusedqk[h] | roped q_pe ]   (computed on the fly)
//   streaming online softmax over causal 32-row key tiles
//   ctx = P @ kva ; out = (ctx / l) @ v_up[h]  -> out_heads[s, h*128 + d]
// ---------------------------------------------------------------------------
__global__ __launch_bounds__(256) void mla_attention(
    const float* __restrict__ qa,    // [S, Q_LORA]  (rmsnorm'd)
    const float* __restrict__ qpe,   // [S, NH*ROPE_D] (roped)
    const float* __restrict__ kf,    // [S, CDIM]
    const float* __restrict__ fqk,   // [NH, Q_LORA, KV_LORA]
    const float* __restrict__ vup,   // [NH, KV_LORA, V_D]
    float* __restrict__ outh)        // [S, NH*V_D]
{
    const int h     = blockIdx.x;
    const int qbase = blockIdx.y * 16;
    const int tid   = threadIdx.x;
    const int w     = tid >> 5;
    const int lane  = tid & 31;
    const int half  = lane >> 4;
    const int khalf = 2 * half;
    const int l16   = lane & 15;
    const int cw    = w * 64;          // this wave's 64-col strip of latent dim

    __shared__ float q_lds[16][584];   // q_full tile; later reused for ctx
    __shared__ float p_lds[16][36];    // logits / probs tile (16x32)
    __shared__ float m_s[16], l_s[16], a_s[16];

    // ---- Step 1: dq tile = q_a[qbase:+16, :] @ fusedqk[h]  -> q_lds[:, 0:512]
    {
        v8f d0 = v8f_zero(), d1 = v8f_zero(), d2 = v8f_zero(), d3 = v8f_zero();
        const float* __restrict__ qar = qa + (size_t)(qbase + l16) * Q_LORA + khalf;
        const float* __restrict__ fb =
            fqk + (size_t)h * Q_LORA * KV_LORA + cw + l16;
        for (int k = 0; k < Q_LORA; k += 4) {
            const v2f a = ld2(qar + k);
            const size_t ro = (size_t)(k + khalf) * KV_LORA;
            v2f b;
            b.x = fb[ro];      b.y = fb[ro + KV_LORA];      d0 = wmma_f32(a, b, d0);
            b.x = fb[ro + 16]; b.y = fb[ro + KV_LORA + 16]; d1 = wmma_f32(a, b, d1);
            b.x = fb[ro + 32]; b.y = fb[ro + KV_LORA + 32]; d2 = wmma_f32(a, b, d2);
            b.x = fb[ro + 48]; b.y = fb[ro + KV_LORA + 48]; d3 = wmma_f32(a, b, d3);
        }
        v8f ds[4] = {d0, d1, d2, d3};
        for (int j = 0; j < 4; ++j)
            for (int r = 0; r < 8; ++r)
                q_lds[r + 8 * half][cw + j * 16 + l16] = ds[j][r];
    }
    // rope part of q into cols 512..575
    for (int idx = tid; idx < 16 * ROPE_D; idx += 256) {
        const int rr = idx >> 6, cc = idx & 63;
        q_lds[rr][KV_LORA + cc] =
            qpe[(size_t)(qbase + rr) * (NH * ROPE_D) + h * ROPE_D + cc];
    }
    if (tid < 16) { m_s[tid] = -3.0e38f; l_s[tid] = 0.f; }
    v8f c0 = v8f_zero(), c1 = v8f_zero(), c2 = v8f_zero(), c3 = v8f_zero();
    __syncthreads();

    // ---- Step 2: stream causal key tiles of 32 rows
    const int ktmax = (qbase + 15) >> 5;    // inclusive
    for (int kt = 0; kt <= ktmax; ++kt) {
        const int kbase = kt << 5;

        // 2a: logits tile [16 x 32]; waves 0,1 compute one 16x16 tile each
        if (w < 2) {
            v8f sa = v8f_zero();
            const float* __restrict__ kr =
                kf + (size_t)(kbase + w * 16 + l16) * CDIM + khalf;
            const float* __restrict__ qrow = &q_lds[l16][khalf];
            for (int c = 0; c < CDIM; c += 4) {
                const v2f a = ld2(qrow + c);
                const v2f b = ld2(kr + c);
                sa = wmma_f32(a, b, sa);
            }
            for (int r = 0; r < 8; ++r) {
                const int rr = r + 8 * half;
                const int cl = w * 16 + l16;
                const float v = ((kbase + cl) > (qbase + rr))
                                    ? MIN_MASKED : sa[r] * SM_SCALE;
                p_lds[rr][cl] = v;
            }
        }
        __syncthreads();

        // 2b: online softmax row update (one thread per row)
        if (tid < 16) {
            const float mo = m_s[tid];
            float mt = mo;
            for (int c = 0; c < 32; ++c) mt = fmaxf(mt, p_lds[tid][c]);
            const float al = expf(mo - mt);
            float sum = 0.f;
            for (int c = 0; c < 32; ++c) {
                const float e = expf(p_lds[tid][c] - mt);
                p_lds[tid][c] = e;
                sum += e;
            }
            m_s[tid] = mt;
            a_s[tid] = al;
            l_s[tid] = al * l_s[tid] + sum;
        }
        __syncthreads();

        // 2c: ctx = alpha*ctx + P @ kva_tile (all 8 waves, 64-col strips)
        {
            const float alr = a_s[8 * half];        // rows r+8*half share... no:
            // per-element row rescale (row = r + 8*half)
            for (int r = 0; r < 8; ++r) {
                const float al = a_s[r + 8 * half];
                c0[r] *= al; c1[r] *= al; c2[r] *= al; c3[r] *= al;
            }
            (void)alr;
            const float* __restrict__ kb = kf + (size_t)kbase * CDIM + cw + l16;
            const float* __restrict__ prow = &p_lds[l16][khalf];
            for (int c = 0; c < 32; c += 4) {
                const v2f a = ld2(prow + c);
                const size_t ro = (size_t)(c + khalf) * CDIM;
                v2f b;
                b.x = kb[ro];      b.y = kb[ro + CDIM];      c0 = wmma_f32(a, b, c0);
                b.x = kb[ro + 16]; b.y = kb[ro + CDIM + 16]; c1 = wmma_f32(a, b, c1);
                b.x = kb[ro + 32]; b.y = kb[ro + CDIM + 32]; c2 = wmma_f32(a, b, c2);
                b.x = kb[ro + 48]; b.y = kb[ro + CDIM + 48]; c3 = wmma_f32(a, b, c3);
            }
        }
        __syncthreads();   // protect p_lds / stats for the next tile
    }

    // ---- Step 3: normalize and stage ctx back into LDS
    if (tid < 16) a_s[tid] = 1.0f / l_s[tid];
    __syncthreads();
    {
        v8f cs[4] = {c0, c1, c2, c3};
        for (int j = 0; j < 4; ++j)
            for (int r = 0; r < 8; ++r)
                q_lds[r + 8 * half][cw + j * 16 + l16] =
                    cs[j][r] * a_s[r + 8 * half];
    }
    __syncthreads();

    // ---- Step 4: out tile [16 x 128] = ctx @ v_up[h]; wave w -> 16-col strip
    v8f oa = v8f_zero();
    const float* __restrict__ vb =
        vup + (size_t)h * KV_LORA * V_D + w * 16 + l16;
    const float* __restrict__ crow = &q_lds[l16][khalf];
    for (int c = 0; c < KV_LORA; c += 4) {
        const v2f a = ld2(crow + c);
        const size_t ro = (size_t)(c + khalf) * V_D;
        v2f b;
        b.x = vb[ro];
        b.y = vb[ro + V_D];
        oa = wmma_f32(a, b, oa);
    }
    for (int r = 0; r < 8; ++r)
        outh[(size_t)(qbase + r + 8 * half) * OUTHW + h * V_D + w * 16 + l16]
            = oa[r];
}

// ---------------------------------------------------------------------------
// Host launcher
// ---------------------------------------------------------------------------
extern "C" void kernel_launch(void* const* d_in, const int* in_sizes, int n_in,
                              void* d_out, int out_size, void* d_ws, size_t ws_size,
                              hipStream_t stream) {
    (void)in_sizes; (void)n_in; (void)out_size; (void)ws_size;

    const float* hidden   = (const float*)d_in[0];
    // d_in[1] = attention_mask (bool, unused — causality recomputed)
    const int*   pos_ids  = (const int*)d_in[2];
    const float* W_kv_a   = (const float*)d_in[3];
    const float* W_q_a    = (const float*)d_in[4];
    const float* q_a_ln_w = (const float*)d_in[5];
    const float* kv_a_ln_w= (const float*)d_in[6];
    const float* q_rope_w = (const float*)d_in[7];
    const float* fusedqk  = (const float*)d_in[8];
    const float* v_up     = (const float*)d_in[9];
    const float* W_o      = (const float*)d_in[10];
    float*       out      = (float*)d_out;

    float* ws = (float*)d_ws;
    float* compressed = ws;                                   // [S, 576]
    float* kfull      = compressed + (size_t)S_LEN * CDIM;    // [S, 576]
    float* q_a        = kfull + (size_t)S_LEN * CDIM;         // [S, 1536]
    float* q_pe       = q_a + (size_t)S_LEN * Q_LORA;         // [S, 2048]
    float* out_heads  = q_pe + (size_t)S_LEN * (NH * ROPE_D); // [S, 4096]

    const dim3 blk(256);

    // 1. compressed = hidden @ W_kv_a   [2048 x 576]
    gemm_wmma_f32<<<dim3((CDIM + 127) / 128, S_LEN / 64), blk, 0, stream>>>(
        hidden, W_kv_a, compressed, S_LEN, CDIM, HID, HID, CDIM, CDIM);

    // 2. k_full = [rmsnorm(kva) | rope(k_pe)]
    build_kfull<<<dim3(S_LEN), blk, 0, stream>>>(compressed, kv_a_ln_w, pos_ids, kfull);

    // 3. q_a = hidden @ W_q_a, then RMSNorm in place
    gemm_wmma_f32<<<dim3((Q_LORA + 127) / 128, S_LEN / 64), blk, 0, stream>>>(
        hidden, W_q_a, q_a, S_LEN, Q_LORA, HID, HID, Q_LORA, Q_LORA);
    rmsnorm_rows<<<dim3(S_LEN), blk, 0, stream>>>(q_a, q_a_ln_w, Q_LORA);

    // 4. q_pe = q_a @ q_rope_w, then RoPE in place
    gemm_wmma_f32<<<dim3((NH * ROPE_D + 127) / 128, S_LEN / 64), blk, 0, stream>>>(
        q_a, q_rope_w, q_pe, S_LEN, NH * ROPE_D, Q_LORA,
        Q_LORA, NH * ROPE_D, NH * ROPE_D);
    rope_qpe<<<dim3(S_LEN), blk, 0, stream>>>(q_pe, pos_ids);

    // 5. fused attention (dq + flash softmax + ctx + v_up) -> out_heads
    mla_attention<<<dim3(NH, S_LEN / 16), blk, 0, stream>>>(
        q_a, q_pe, kfull, fusedqk, v_up, out_heads);

    // 6. out = out_heads @ W_o   [2048 x 7168]
    gemm_wmma_f32<<<dim3((HID + 127) / 128, S_LEN / 64), blk, 0, stream>>>(
        out_heads, W_o, out, S_LEN, HID, OUTHW, OUTHW, HID, HID);
}